// SelfAttentionHead_33543694582124
// MI455X (gfx1250) — compile-verified
//
#include <hip/hip_runtime.h>
#include <math.h>

typedef __attribute__((ext_vector_type(2))) float v2f;
typedef __attribute__((ext_vector_type(8))) float v8f;

#define D_MODEL 1024
#define SEQ_B 4
#define SEQ_T 2048

__device__ __forceinline__ v8f wmma4(v2f a, v2f b, v8f c) {
  // D = A(16x4, f32) x B(4x16, f32) + C(16x16, f32)
  return __builtin_amdgcn_wmma_f32_16x16x4_f32(false, a, false, b, (short)0, c,
                                               false, false);
}

// Async global->LDS copy of one 16B chunk per lane (CDNA5, ASYNCcnt-tracked).
__device__ __forceinline__ void async_copy_b128(unsigned lds_off,
                                                const void* gptr) {
  unsigned long long ga = (unsigned long long)(uintptr_t)gptr;
  asm volatile("global_load_async_to_lds_b128 %0, %1, off"
               :
               : "v"(lds_off), "v"(ga)
               : "memory");
}
__device__ __forceinline__ void wait_async() {
  asm volatile("s_wait_asynccnt 0x0" ::: "memory");
}

// ---------------------------------------------------------------------------
// Generic tiled GEMM: C[M,N] = act(A[M,K] @ W[K,N] + bias) [+ res]
// 256 threads = 8 waves; block tile 64x128; wave tile 32x32 (2x2 WMMA accums)
// A tile [64][32] and W tile [32][128] staged via async global->LDS DMA.
// ---------------------------------------------------------------------------
__global__ __launch_bounds__(256) void gemm_bias_kernel(
    const float* __restrict__ A, const float* __restrict__ W,
    const float* __restrict__ bias, const float* __restrict__ res,
    float* __restrict__ C, int M, int N, int K, int relu) {
  __shared__ float As[64 * 32];   // [row][k] row-major
  __shared__ float Ws[32 * 128];  // [k][n] natural layout (async-friendly)

  const int tid = threadIdx.x;
  const int lane = tid & 31;
  const int wid = tid >> 5;
  const int laneLo = lane & 15, laneHi = lane >> 4;
  const int wm = wid >> 2, wn = wid & 3;  // 2x4 wave grid
  const int m0 = blockIdx.y * 64, n0 = blockIdx.x * 128;

  // Per-thread staging coordinates (constant across k-chunks)
  const int arow = tid >> 3, ac4 = tid & 7;          // A: 2 chunks/thread
  const int wkr = tid >> 5, wc4 = tid & 31;          // W: 4 chunks/thread
  const unsigned asOff0 = (unsigned)(uintptr_t)&As[(arow)*32 + ac4 * 4];
  const unsigned asOff1 = (unsigned)(uintptr_t)&As[(arow + 32) * 32 + ac4 * 4];
  unsigned wsOff[4];
#pragma unroll
  for (int i = 0; i < 4; i++)
    wsOff[i] = (unsigned)(uintptr_t)&Ws[(wkr + i * 8) * 128 + wc4 * 4];

  v8f acc[2][2];
  for (int i = 0; i < 2; i++)
    for (int j = 0; j < 2; j++)
      for (int r = 0; r < 8; r++) acc[i][j][r] = 0.f;

  for (int k0 = 0; k0 < K; k0 += 32) {
    // ---- async stage A tile 64x32 (2 x 16B per thread) ----
    async_copy_b128(asOff0, &A[(size_t)(m0 + arow) * K + k0 + ac4 * 4]);
    async_copy_b128(asOff1, &A[(size_t)(m0 + arow + 32) * K + k0 + ac4 * 4]);
    // ---- async stage W tile 32x128 (4 x 16B per thread) ----
#pragma unroll
    for (int i = 0; i < 4; i++)
      async_copy_b128(wsOff[i],
                      &W[(size_t)(k0 + wkr + i * 8) * N + n0 + wc4 * 4]);
    wait_async();
    __syncthreads();

#pragma unroll
    for (int kk = 0; kk < 32; kk += 4) {
      v2f a[2], b[2];
#pragma unroll
      for (int mi = 0; mi < 2; mi++)
        a[mi] =
            *(const v2f*)&As[(wm * 32 + mi * 16 + laneLo) * 32 + kk + 2 * laneHi];
#pragma unroll
      for (int ni = 0; ni < 2; ni++) {
        int nc = wn * 32 + ni * 16 + laneLo;
        v2f bb;
        bb.x = Ws[(kk + 2 * laneHi) * 128 + nc];
        bb.y = Ws[(kk + 2 * laneHi + 1) * 128 + nc];
        b[ni] = bb;
      }
#pragma unroll
      for (int mi = 0; mi < 2; mi++)
#pragma unroll
        for (int ni = 0; ni < 2; ni++)
          acc[mi][ni] = wmma4(a[mi], b[ni], acc[mi][ni]);
    }
    __syncthreads();
  }

  // epilogue: bias, optional relu, optional residual add
#pragma unroll
  for (int mi = 0; mi < 2; mi++) {
#pragma unroll
    for (int ni = 0; ni < 2; ni++) {
      int cg = n0 + wn * 32 + ni * 16 + laneLo;
      float bv = bias[cg];
#pragma unroll
      for (int r = 0; r < 8; r++) {
        int rg = m0 + wm * 32 + mi * 16 + r + 8 * laneHi;
        float v = acc[mi][ni][r] + bv;
        if (relu) v = v > 0.f ? v : 0.f;
        if (res) v += res[(size_t)rg * N + cg];
        C[(size_t)rg * N + cg] = v;
      }
    }
  }
}

// ---------------------------------------------------------------------------
// Fused causal flash-attention (single head, d=1024) + residual + LayerNorm1.
// One block per 16-query tile; each of 8 waves owns a 128-wide d-chunk.
// ---------------------------------------------------------------------------
__global__ __launch_bounds__(256) void attn_ln_kernel(
    const float* __restrict__ Q, const float* __restrict__ Km,
    const float* __restrict__ Vm, const float* __restrict__ X,
    const float* __restrict__ g1, const float* __restrict__ be1,
    float* __restrict__ ALN) {
  __shared__ float red[8 * 256];   // cross-wave partial-score reduction
  __shared__ float Pbuf[8 * 256];  // per-wave P tile (row-major 16x16)
  __shared__ float sbuf[8 * 16], qbuf[8 * 16];
  __shared__ float meanb[16], rstdb[16];

  const int tid = threadIdx.x, lane = tid & 31, wid = tid >> 5;
  const int laneLo = lane & 15, laneHi = lane >> 4;
  const int T = SEQ_T, d = D_MODEL;
  const int qt = blockIdx.x & (T / 16 - 1);
  const int b = blockIdx.x / (T / 16);
  const int q0 = qt * 16;
  const int dbase = wid * 128;
  const size_t base = (size_t)b * T * d;
  const float scale = 0.03125f;  // 1/sqrt(1024)
  const float NEG = -1.0e30f;

  // Q fragments for this wave's d-chunk held in registers for the whole loop
  v2f qf[32];
#pragma unroll
  for (int j = 0; j < 32; j++) {
    int k = dbase + 4 * j + 2 * laneHi;
    qf[j] = *(const v2f*)&Q[base + (size_t)(q0 + laneLo) * d + k];
  }

  v8f o[8];
  for (int t = 0; t < 8; t++)
    for (int r = 0; r < 8; r++) o[t][r] = 0.f;
  float mrow[8], lrow[8];
  for (int r = 0; r < 8; r++) { mrow[r] = NEG; lrow[r] = 0.f; }

  for (int st = 0; st <= qt; st++) {
    const int s0 = st * 16;
    // partial S = Q Kt over this wave's d-chunk (32 WMMAs)
    v8f s;
    for (int r = 0; r < 8; r++) s[r] = 0.f;
#pragma unroll
    for (int j = 0; j < 32; j++) {
      int k = dbase + 4 * j + 2 * laneHi;
      v2f kf = *(const v2f*)&Km[base + (size_t)(s0 + laneLo) * d + k];
      s = wmma4(qf[j], kf, s);
    }
#pragma unroll
    for (int r = 0; r < 8; r++) red[wid * 256 + r * 32 + lane] = s[r];
    __syncthreads();
    float sv[8];
#pragma unroll
    for (int r = 0; r < 8; r++) {
      float a = 0.f;
#pragma unroll
      for (int w = 0; w < 8; w++) a += red[w * 256 + r * 32 + lane];
      sv[r] = a;
    }
    // causal mask + online softmax (row lives across 16 lanes at one vgpr)
    float corr[8];
#pragma unroll
    for (int r = 0; r < 8; r++) {
      int row = r + 8 * laneHi, col = laneLo;
      float val = sv[r] * scale;
      if (s0 + col > q0 + row) val = NEG;
      float rmax = val;
      rmax = fmaxf(rmax, __shfl_xor(rmax, 1, 32));
      rmax = fmaxf(rmax, __shfl_xor(rmax, 2, 32));
      rmax = fmaxf(rmax, __shfl_xor(rmax, 4, 32));
      rmax = fmaxf(rmax, __shfl_xor(rmax, 8, 32));
      float nm = fmaxf(mrow[r], rmax);
      float p = __expf(val - nm);
      float rs = p;
      rs += __shfl_xor(rs, 1, 32);
      rs += __shfl_xor(rs, 2, 32);
      rs += __shfl_xor(rs, 4, 32);
      rs += __shfl_xor(rs, 8, 32);
      float c = __expf(mrow[r] - nm);
      lrow[r] = lrow[r] * c + rs;
      mrow[r] = nm;
      corr[r] = c;
      Pbuf[wid * 256 + row * 16 + col] = p;
    }
#pragma unroll
    for (int t = 0; t < 8; t++)
#pragma unroll
      for (int r = 0; r < 8; r++) o[t][r] *= corr[r];
    __syncthreads();
    // O += P @ V over this wave's d-chunk (32 WMMAs)
#pragma unroll
    for (int kc = 0; kc < 16; kc += 4) {
      v2f af = *(const v2f*)&Pbuf[wid * 256 + laneLo * 16 + kc + 2 * laneHi];
#pragma unroll
      for (int t = 0; t < 8; t++) {
        int nc = dbase + t * 16 + laneLo;
        v2f bf;
        bf.x = Vm[base + (size_t)(s0 + kc + 2 * laneHi) * d + nc];
        bf.y = Vm[base + (size_t)(s0 + kc + 2 * laneHi + 1) * d + nc];
        o[t] = wmma4(af, bf, o[t]);
      }
    }
    __syncthreads();
  }

  // finalize softmax, add residual x
#pragma unroll
  for (int t = 0; t < 8; t++) {
    int nc = dbase + t * 16 + laneLo;
#pragma unroll
    for (int r = 0; r < 8; r++) {
      int row = r + 8 * laneHi;
      o[t][r] = o[t][r] / lrow[r] + X[base + (size_t)(q0 + row) * d + nc];
    }
  }
  // LayerNorm over d=1024 (cross-wave reduction via LDS)
  float ps[8], pq[8];
  for (int r = 0; r < 8; r++) { ps[r] = 0.f; pq[r] = 0.f; }
  for (int t = 0; t < 8; t++)
    for (int r = 0; r < 8; r++) { float y = o[t][r]; ps[r] += y; pq[r] += y * y; }
  for (int r = 0; r < 8; r++)
    for (int m = 1; m <= 8; m <<= 1) {
      ps[r] += __shfl_xor(ps[r], m, 32);
      pq[r] += __shfl_xor(pq[r], m, 32);
    }
  if (laneLo == 0)
    for (int r = 0; r < 8; r++) {
      int row = r + 8 * laneHi;
      sbuf[wid * 16 + row] = ps[r];
      qbuf[wid * 16 + row] = pq[r];
    }
  __syncthreads();
  if (tid < 16) {
    float S = 0.f, Q2 = 0.f;
    for (int w = 0; w < 8; w++) { S += sbuf[w * 16 + tid]; Q2 += qbuf[w * 16 + tid]; }
    float mean = S / (float)d;
    meanb[tid] = mean;
    rstdb[tid] = rsqrtf(Q2 / (float)d - mean * mean + 1e-5f);
  }
  __syncthreads();
  for (int t = 0; t < 8; t++) {
    int nc = dbase + t * 16 + laneLo;
    float gg = g1[nc], bb = be1[nc];
    for (int r = 0; r < 8; r++) {
      int row = r + 8 * laneHi;
      ALN[base + (size_t)(q0 + row) * d + nc] =
          (o[t][r] - meanb[row]) * rstdb[row] * gg + bb;
    }
  }
}

// ---------------------------------------------------------------------------
// In-place LayerNorm over d=1024: one block per row.
// ---------------------------------------------------------------------------
__global__ __launch_bounds__(256) void ln_kernel(float* __restrict__ Y,
                                                 const float* __restrict__ g,
                                                 const float* __restrict__ b,
                                                 int d) {
  __shared__ float sb[8], qb[8];
  const int row = blockIdx.x;
  const int tid = threadIdx.x, lane = tid & 31, wid = tid >> 5;
  float v[4];
  float s = 0.f, q = 0.f;
  for (int i = 0; i < 4; i++) {
    v[i] = Y[(size_t)row * d + tid + i * 256];
    s += v[i];
    q += v[i] * v[i];
  }
  for (int m = 1; m < 32; m <<= 1) {
    s += __shfl_xor(s, m, 32);
    q += __shfl_xor(q, m, 32);
  }
  if (lane == 0) { sb[wid] = s; qb[wid] = q; }
  __syncthreads();
  if (tid == 0) {
    float S = 0.f, Q2 = 0.f;
    for (int w = 0; w < 8; w++) { S += sb[w]; Q2 += qb[w]; }
    float mean = S / (float)d;
    sb[0] = mean;
    qb[0] = rsqrtf(Q2 / (float)d - mean * mean + 1e-5f);
  }
  __syncthreads();
  float mean = sb[0], rstd = qb[0];
  for (int i = 0; i < 4; i++) {
    int c = tid + i * 256;
    Y[(size_t)row * d + c] = (v[i] - mean) * rstd * g[c] + b[c];
  }
}

// ---------------------------------------------------------------------------
extern "C" void kernel_launch(void* const* d_in, const int* in_sizes, int n_in,
                              void* d_out, int out_size, void* d_ws,
                              size_t ws_size, hipStream_t stream) {
  (void)in_sizes; (void)n_in; (void)out_size; (void)ws_size;
  const float* x   = (const float*)d_in[0];
  const float* Wq  = (const float*)d_in[1];
  const float* bq  = (const float*)d_in[2];
  const float* Wk  = (const float*)d_in[3];
  const float* bk  = (const float*)d_in[4];
  const float* Wv  = (const float*)d_in[5];
  const float* bv  = (const float*)d_in[6];
  const float* g1  = (const float*)d_in[7];
  const float* be1 = (const float*)d_in[8];
  const float* W1  = (const float*)d_in[9];
  const float* b1  = (const float*)d_in[10];
  const float* W2  = (const float*)d_in[11];
  const float* b2  = (const float*)d_in[12];
  const float* g2  = (const float*)d_in[13];
  const float* be2 = (const float*)d_in[14];
  float* out = (float*)d_out;

  const int M = SEQ_B * SEQ_T;  // 8192 tokens
  const int d = D_MODEL;        // 1024
  float* ws = (float*)d_ws;
  float* Qb = ws;                         // [0, 8M) floats
  float* Kb = ws + (size_t)M * d;         // [8M, 16M)
  float* Vb = ws + (size_t)2 * M * d;     // [16M, 24M)
  float* H  = ws;                         // [0, 32M) — reuses Q/K/V after attn
  float* ALN = ws + (size_t)4 * M * d;    // [32M, 40M)

  dim3 blk(256);
  gemm_bias_kernel<<<dim3(d / 128, M / 64), blk, 0, stream>>>(
      x, Wq, bq, nullptr, Qb, M, d, d, 0);
  gemm_bias_kernel<<<dim3(d / 128, M / 64), blk, 0, stream>>>(
      x, Wk, bk, nullptr, Kb, M, d, d, 0);
  gemm_bias_kernel<<<dim3(d / 128, M / 64), blk, 0, stream>>>(
      x, Wv, bv, nullptr, Vb, M, d, d, 0);
  attn_ln_kernel<<<dim3(M / 16), blk, 0, stream>>>(Qb, Kb, Vb, x, g1, be1, ALN);
  gemm_bias_kernel<<<dim3(4 * d / 128, M / 64), blk, 0, stream>>>(
      ALN, W1, b1, nullptr, H, M, 4 * d, d, 1);
  gemm_bias_kernel<<<dim3(d / 128, M / 64), blk, 0, stream>>>(
      H, W2, b2, ALN, out, M, d, 4 * d, 0);
  ln_kernel<<<dim3(M), blk, 0, stream>>>(out, g2, be2, d);
}